// GatLayer_35562329210937
// MI455X (gfx1250) — compile-verified
//
#include <hip/hip_runtime.h>
#include <hip/hip_bf16.h>
#include <math.h>

typedef float v2f __attribute__((ext_vector_type(2)));
typedef float v8f __attribute__((ext_vector_type(8)));

#define N_NODES 4096
#define B_SZ    4
#define F_SZ    64
#define ALPHA   0.2f
#define NEG_BIG -9e15f
#define L2E     1.4426950408889634f

// ---------------------------------------------------------------------------
// Kernel 1: Wh = h @ W (per node), stored PERMUTED for the WMMA B-operand:
//   WhP[j][ (o%16)*4 + (o/16) ] = Wh[j][o]
// so a lane can fetch its 4 o-block values for one row as one b128 load.
// Also s' = (Wh·a1)/64, t' = (Wh·a2)/64  (dk = sqrt(4096) = 64, exact /64).
// ---------------------------------------------------------------------------
__global__ __launch_bounds__(64) void wh_kernel(
    const float* __restrict__ h, const float* __restrict__ W,
    const float* __restrict__ a, float* __restrict__ WhP,
    float* __restrict__ s1, float* __restrict__ t2)
{
    int bi = blockIdx.x;             // b*4096 + i
    int o  = threadIdx.x;            // 0..63
    const float* hrow = h + (size_t)bi * F_SZ;
    float acc = 0.f;
#pragma unroll
    for (int f = 0; f < F_SZ; ++f)
        acc = fmaf(hrow[f], W[f * F_SZ + o], acc);
    int operm = ((o & 15) << 2) | (o >> 4);
    WhP[(size_t)bi * F_SZ + operm] = acc;

    __shared__ float sWh[F_SZ];
    sWh[o] = acc;
    __syncthreads();
    if (o == 0) {
        float s = 0.f;
        for (int k = 0; k < F_SZ; ++k) s = fmaf(sWh[k], a[k], s);
        s1[bi] = s * (1.0f / 64.0f);
    } else if (o == 1) {
        float t = 0.f;
        for (int k = 0; k < F_SZ; ++k) t = fmaf(sWh[k], a[F_SZ + k], t);
        t2[bi] = t * (1.0f / 64.0f);
    }
}

// ---------------------------------------------------------------------------
// Kernel 2: masked row max. lrelu((s+t)/dk) is monotone in t, so
// m[b,i] = hasNbr ? lrelu(s'[b,i] + max_{j:adj>0} t'[b,j]) : -9e15.
// One wave per node; adj row scanned once for all 4 batches.
// ---------------------------------------------------------------------------
__global__ __launch_bounds__(256) void rowmax_kernel(
    const int* __restrict__ adj, const float* __restrict__ s1,
    const float* __restrict__ t2, float* __restrict__ mrow)
{
    int wave = threadIdx.x >> 5;
    int lane = threadIdx.x & 31;
    int i = blockIdx.x * 8 + wave;
    const int* arow = adj + (size_t)i * N_NODES;
    float T0 = -3.0e38f, T1 = -3.0e38f, T2 = -3.0e38f, T3 = -3.0e38f;
    int any = 0;
    for (int j = lane; j < N_NODES; j += 32) {
        int av = arow[j];
        bool sel = av > 0;
        any |= (int)sel;
        float u0 = t2[j];
        float u1 = t2[N_NODES + j];
        float u2 = t2[2 * N_NODES + j];
        float u3 = t2[3 * N_NODES + j];
        T0 = fmaxf(T0, sel ? u0 : -3.0e38f);
        T1 = fmaxf(T1, sel ? u1 : -3.0e38f);
        T2 = fmaxf(T2, sel ? u2 : -3.0e38f);
        T3 = fmaxf(T3, sel ? u3 : -3.0e38f);
    }
#pragma unroll
    for (int off = 16; off >= 1; off >>= 1) {
        T0 = fmaxf(T0, __shfl_xor(T0, off, 32));
        T1 = fmaxf(T1, __shfl_xor(T1, off, 32));
        T2 = fmaxf(T2, __shfl_xor(T2, off, 32));
        T3 = fmaxf(T3, __shfl_xor(T3, off, 32));
        any |= __shfl_xor(any, off, 32);
    }
    if (lane == 0) {
        float Tb[4] = {T0, T1, T2, T3};
#pragma unroll
        for (int b = 0; b < 4; ++b) {
            float x = s1[b * N_NODES + i] + Tb[b];
            float m = any ? fmaxf(x, ALPHA * x) : NEG_BIG;
            mrow[b * N_NODES + i] = m;
        }
    }
}

// ---------------------------------------------------------------------------
// Kernel 3: fused score-gen + P@Wh via V_WMMA_F32_16X16X4_F32.
// One wave (32 lanes) per (chunk, b, 16-row tile).
// A-layout (f32 16x4): lane = M%16; lanes 0-15 hold K=0,1; lanes 16-31 K=2,3.
// B tiles come from the permuted WhP: one b128 per K-row per lane.
// Output stored in the same o-permuted order (bn_kernel undoes it).
// ---------------------------------------------------------------------------
__global__ __launch_bounds__(32) void attn_kernel(
    const int* __restrict__ adj, const float* __restrict__ WhP,
    const float* __restrict__ s1, const float* __restrict__ t2,
    const float* __restrict__ mrow, float* __restrict__ Dp,
    float* __restrict__ Zp, int jspan)
{
    int bid   = blockIdx.x;
    int b     = bid & 3;
    int itile = (bid >> 2) & 255;
    int chunk = bid >> 10;
    int lane  = threadIdx.x;
    int l15   = lane & 15;
    int half  = lane >> 4;
    int irow  = itile * 16 + l15;

    float s = s1[b * N_NODES + irow];
    float m = mrow[b * N_NODES + irow];
    const int*   adjrow = adj + (size_t)irow * N_NODES;
    const float* trow   = t2 + (size_t)b * N_NODES;
    const float* WhB    = WhP + (size_t)b * N_NODES * F_SZ;

    v8f c0 = {}, c1 = {}, c2 = {}, c3 = {};
    float zacc = 0.f;

    int j0 = chunk * jspan;
    int jend = j0 + jspan;
#pragma unroll 4
    for (int j = j0; j < jend; j += 4) {
        int jj = j + half * 2;
        int2   av = *(const int2*)(adjrow + jj);
        float2 tv = *(const float2*)(trow + jj);

        float x0 = s + tv.x, x1 = s + tv.y;
        float e0 = fmaxf(x0, ALPHA * x0);          // leaky relu
        float e1 = fmaxf(x1, ALPHA * x1);
        float sc0 = (av.x > 0) ? e0 : NEG_BIG;
        float sc1 = (av.y > 0) ? e1 : NEG_BIG;
        float p0 = __builtin_amdgcn_exp2f((sc0 - m) * L2E);
        float p1 = __builtin_amdgcn_exp2f((sc1 - m) * L2E);
        zacc += p0 + p1;

        v2f A; A[0] = p0; A[1] = p1;

        // permuted B fetch: one float4 per K-row covering all 4 o-blocks
        const float* bp = WhB + (size_t)jj * F_SZ + (l15 << 2);
        float4 q0 = *(const float4*)(bp);
        float4 q1 = *(const float4*)(bp + F_SZ);
        v2f B0, B1, B2, B3;
        B0[0] = q0.x; B0[1] = q1.x;
        B1[0] = q0.y; B1[1] = q1.y;
        B2[0] = q0.z; B2[1] = q1.z;
        B3[0] = q0.w; B3[1] = q1.w;

        c0 = __builtin_amdgcn_wmma_f32_16x16x4_f32(false, A, false, B0, (short)0, c0, false, false);
        c1 = __builtin_amdgcn_wmma_f32_16x16x4_f32(false, A, false, B1, (short)0, c1, false, false);
        c2 = __builtin_amdgcn_wmma_f32_16x16x4_f32(false, A, false, B2, (short)0, c2, false, false);
        c3 = __builtin_amdgcn_wmma_f32_16x16x4_f32(false, A, false, B3, (short)0, c3, false, false);
    }

    // combine Z halves (K=0,1 partial lives in lane i, K=2,3 in lane i+16)
    zacc += __shfl_xor(zacc, 16, 32);

    // C/D layout: vgpr r -> (row r, col lane) for lanes 0-15; (row r+8, col lane-16)
    // store permuted: Dp[row][l15*4 + blk]
    size_t obase = (((size_t)chunk * 4 + b) * N_NODES + (size_t)itile * 16) * F_SZ;
    float* dout = Dp + obase;
    int rbase = half * 8;
#pragma unroll
    for (int r = 0; r < 8; ++r) {
        float4 v; v.x = c0[r]; v.y = c1[r]; v.z = c2[r]; v.w = c3[r];
        *(float4*)(dout + (size_t)(rbase + r) * F_SZ + (l15 << 2)) = v;
    }
    if (half == 0)
        Zp[((size_t)chunk * 4 + b) * N_NODES + irow] = zacc;
}

// ---------------------------------------------------------------------------
// Kernel 4: split-K combine + divide by Z + BatchNorm over (B,Fout) per node
// + residual + ELU. One block (256 = 4*64 threads) per node.
// Dp is stored o-permuted: element o lives at (o%16)*4 + o/16.
// ---------------------------------------------------------------------------
__global__ __launch_bounds__(256) void bn_kernel(
    const float* __restrict__ Dp, const float* __restrict__ Zp,
    const float* __restrict__ h, const float* __restrict__ gamma,
    const float* __restrict__ beta, float* __restrict__ out, int nchunk)
{
    int i   = blockIdx.x;
    int tid = threadIdx.x;
    int b   = tid >> 6;
    int o   = tid & 63;
    int operm = ((o & 15) << 2) | (o >> 4);

    float num = 0.f, z = 0.f;
    for (int c = 0; c < nchunk; ++c) {
        num += Dp[((((size_t)c * 4 + b) * N_NODES + i) << 6) + operm];
        z   += Zp[(((size_t)c * 4 + b) * N_NODES) + i];
    }
    float hp = num / z;

    __shared__ float red[256];
    red[tid] = hp;
    __syncthreads();
#pragma unroll
    for (int s = 128; s >= 1; s >>= 1) {
        if (tid < s) red[tid] += red[tid + s];
        __syncthreads();
    }
    float mean = red[0] * (1.0f / 256.0f);
    __syncthreads();
    float dev = hp - mean;
    red[tid] = dev * dev;
    __syncthreads();
#pragma unroll
    for (int s = 128; s >= 1; s >>= 1) {
        if (tid < s) red[tid] += red[tid + s];
        __syncthreads();
    }
    float var = red[0] * (1.0f / 256.0f);

    float xn  = gamma[i] * dev * (1.0f / sqrtf(var + 1e-5f)) + beta[i];
    float res = xn + h[((size_t)b * N_NODES + i) * F_SZ + o];
    out[((size_t)b * N_NODES + i) * F_SZ + o] = res > 0.f ? res : expm1f(res);
}

// ---------------------------------------------------------------------------
extern "C" void kernel_launch(void* const* d_in, const int* in_sizes, int n_in,
                              void* d_out, int out_size, void* d_ws, size_t ws_size,
                              hipStream_t stream)
{
    const float* h     = (const float*)d_in[0];   // [4,4096,64]
    const int*   adj   = (const int*)  d_in[1];   // [4096,4096]
    const float* W     = (const float*)d_in[2];   // [64,64]
    const float* a     = (const float*)d_in[3];   // [128]
    const float* gamma = (const float*)d_in[4];   // [4096]
    const float* beta  = (const float*)d_in[5];   // [4096]
    float* out = (float*)d_out;

    const size_t WH_ELEMS = (size_t)B_SZ * N_NODES * F_SZ;   // 1,048,576
    const size_t NROW     = (size_t)B_SZ * N_NODES;          // 16,384

    // pick split-K factor based on workspace size
    int nchunk = 4;
    size_t need4 = (WH_ELEMS + 3 * NROW + 4 * (WH_ELEMS + NROW)) * sizeof(float);
    if (ws_size < need4) nchunk = 1;

    float* wsf  = (float*)d_ws;
    float* WhP  = wsf;
    float* s1   = WhP + WH_ELEMS;
    float* t2   = s1 + NROW;
    float* mrow = t2 + NROW;
    float* Dp   = mrow + NROW;
    float* Zp   = Dp + (size_t)nchunk * WH_ELEMS;

    int jspan = N_NODES / nchunk;

    wh_kernel<<<B_SZ * N_NODES, 64, 0, stream>>>(h, W, a, WhP, s1, t2);
    rowmax_kernel<<<N_NODES / 8, 256, 0, stream>>>(adj, s1, t2, mrow);
    attn_kernel<<<B_SZ * (N_NODES / 16) * nchunk, 32, 0, stream>>>(
        adj, WhP, s1, t2, mrow, Dp, Zp, jspan);
    bn_kernel<<<N_NODES, 256, 0, stream>>>(Dp, Zp, h, gamma, beta, out, nchunk);
}